// EGNNGlobalNodeHetero_38019050504306
// MI455X (gfx1250) — compile-verified
//
#include <hip/hip_runtime.h>

typedef float v2f __attribute__((ext_vector_type(2)));
typedef float v8f __attribute__((ext_vector_type(8)));

#define TPB 256
#define GRID_BIG 448
#define LN_BLOCKS 512

// ---------------------------------------------------------------- device utils
__device__ __forceinline__ v8f wmma4(v2f a, v2f b, v8f c) {
  // D = A(16x4,f32) * B(4x16,f32) + C(16x16,f32)
  return __builtin_amdgcn_wmma_f32_16x16x4_f32(false, a, false, b, (short)0, c,
                                               false, false);
}
__device__ __forceinline__ float siluf(float x) { return x / (1.f + __expf(-x)); }
__device__ __forceinline__ float sigmf(float x) { return 1.f / (1.f + __expf(-x)); }
// sum across the 16 lanes of each half-wave (rows live per half-wave)
__device__ __forceinline__ float red16(float v) {
  v += __shfl_xor(v, 1, 32);
  v += __shfl_xor(v, 2, 32);
  v += __shfl_xor(v, 4, 32);
  v += __shfl_xor(v, 8, 32);
  return v;
}
// Stage a K x 128 row-major weight matrix into LDS with (k,k+1) pairs adjacent:
// lds[(k>>1)*256 + n*2 + (k&1)] = W[k*128+n]  -> B fragment is one b64 LDS load,
// and adjacent n-tiles pack into ds_load_2addr_b64.
__device__ __forceinline__ void stage_w(const float* W, float* lds, int K) {
  for (int i = threadIdx.x; i < K * 128; i += blockDim.x) {
    int k = i >> 7, n = i & 127;
    lds[((k >> 1) << 8) + (n << 1) + (k & 1)] = W[i];
  }
}
__device__ __forceinline__ void stage_v(const float* s, float* lds, int n) {
  for (int i = threadIdx.x; i < n; i += blockDim.x) lds[i] = s[i];
}

// One K=128 GEMM sweep: acc(16x128) += row-tile(16x128) @ W(128x128).
// Each lane supplies the A fragment for row m=lane&15, k-pair offset 2*(lane>>4).
// Loads of the 8 B fragments are batched ahead of the 8 WMMAs so a single
// s_wait_dscnt gates an 8-deep back-to-back WMMA burst.
template <bool NORM>
__device__ __forceinline__ void kloop128(const float* row, const float* Wl,
                                         int lane, float mean, float rstd,
                                         const float* lnw, const float* lnb,
                                         v8f acc[8]) {
  const int n = lane & 15;
  const int koff = (lane >> 4) << 1;
  for (int k0 = 0; k0 < 128; k0 += 4) {
    const int kk = k0 + koff;
    v2f a = *(const v2f*)(row + kk);
    if (NORM) {
      a.x = (a.x - mean) * rstd * lnw[kk] + lnb[kk];
      a.y = (a.y - mean) * rstd * lnw[kk + 1] + lnb[kk + 1];
    }
    const float* bp = Wl + ((kk >> 1) << 8) + (n << 1);
    v2f b[8];
#pragma unroll
    for (int t = 0; t < 8; ++t) b[t] = *(const v2f*)(bp + t * 32);
#pragma unroll
    for (int t = 0; t < 8; ++t) acc[t] = wmma4(a, b[t], acc[t]);
  }
}

// ---------------------------------------------------------------- LN statistics
// graph-LN normalizes the *gathered* tensors x_dst[dst] and x_src[src] over all
// E*128 elements. Two-stage deterministic f64 reduction.
__global__ void k_ln_partial(const float* __restrict__ xd, const int* __restrict__ dst,
                             const float* __restrict__ xs, const int* __restrict__ src,
                             int E, double* __restrict__ part) {
  __shared__ double sm[4][TPB];
  double si = 0, qi = 0, sj = 0, qj = 0;
  const long long total = (long long)E * 128;
  const long long stride = (long long)gridDim.x * blockDim.x;
  for (long long i = (long long)blockIdx.x * blockDim.x + threadIdx.x; i < total; i += stride) {
    int e = (int)(i >> 7), f = (int)(i & 127);
    float vi = xd[(size_t)dst[e] * 128 + f];
    float vj = xs[(size_t)src[e] * 128 + f];
    si += vi; qi += (double)vi * vi;
    sj += vj; qj += (double)vj * vj;
  }
  sm[0][threadIdx.x] = si; sm[1][threadIdx.x] = qi;
  sm[2][threadIdx.x] = sj; sm[3][threadIdx.x] = qj;
  __syncthreads();
  for (int off = TPB / 2; off > 0; off >>= 1) {
    if (threadIdx.x < off)
      for (int c = 0; c < 4; ++c) sm[c][threadIdx.x] += sm[c][threadIdx.x + off];
    __syncthreads();
  }
  if (threadIdx.x == 0)
    for (int c = 0; c < 4; ++c) part[blockIdx.x * 4 + c] = sm[c][0];
}

__global__ void k_ln_final(const double* __restrict__ part, int np, long long N,
                           float* __restrict__ stats) {
  __shared__ double sm[4][TPB];
  double a = 0, b = 0, c = 0, d = 0;
  for (int i = threadIdx.x; i < np; i += blockDim.x) {
    a += part[i * 4 + 0]; b += part[i * 4 + 1];
    c += part[i * 4 + 2]; d += part[i * 4 + 3];
  }
  sm[0][threadIdx.x] = a; sm[1][threadIdx.x] = b;
  sm[2][threadIdx.x] = c; sm[3][threadIdx.x] = d;
  __syncthreads();
  for (int off = TPB / 2; off > 0; off >>= 1) {
    if (threadIdx.x < off)
      for (int k = 0; k < 4; ++k) sm[k][threadIdx.x] += sm[k][threadIdx.x + off];
    __syncthreads();
  }
  if (threadIdx.x == 0) {
    double inv = 1.0 / (double)N;
    double mi = sm[0][0] * inv, vi = sm[1][0] * inv - mi * mi;
    double mj = sm[2][0] * inv, vj = sm[3][0] * inv - mj * mj;
    stats[0] = (float)mi; stats[1] = (float)(1.0 / sqrt(vi + 1e-5));
    stats[2] = (float)mj; stats[3] = (float)(1.0 / sqrt(vj + 1e-5));
  }
}

// ---------------------------------------------------------------- edge m1
// h = silu([LN(x_i)|LN(x_j)|dist] @ W1 + b1); dist column handled as rank-1.
template <bool NORM>
__global__ void k_edge_m1(const float* xsrc, const float* xdst,
                          const float* psrc, const float* pdst,
                          const int* __restrict__ src, const int* __restrict__ dst,
                          int E, const float* __restrict__ W1,
                          const float* __restrict__ b1, const float* __restrict__ lnw,
                          const float* __restrict__ lnb, const float* __restrict__ stats,
                          float* __restrict__ hout) {
  extern __shared__ float lds[];
  float* Wa = lds;               // 16384 floats (rows 0..127, x_i part)
  float* Wb = lds + 16384;       // 16384 floats (rows 128..255, x_j part)
  float* cb = lds + 32768;       // b1 | w1c | lnw | lnb  (4*128)
  stage_w(W1, Wa, 128);
  stage_w(W1 + 128 * 128, Wb, 128);
  stage_v(b1, cb, 128);
  stage_v(W1 + 256 * 128, cb + 128, 128);
  if (NORM) { stage_v(lnw, cb + 256, 128); stage_v(lnb, cb + 384, 128); }
  __syncthreads();
  float mi = 0.f, ri = 1.f, mj = 0.f, rj = 1.f;
  if (NORM) { mi = stats[0]; ri = stats[1]; mj = stats[2]; rj = stats[3]; }
  const int lane = threadIdx.x & 31, hi = lane >> 4, m = lane & 15;
  const int wid = (blockIdx.x * blockDim.x + threadIdx.x) >> 5;
  const int nw = (gridDim.x * blockDim.x) >> 5;
  const int nTiles = (E + 15) >> 4;
  for (int tile = wid; tile < nTiles; tile += nw) {
    int e = tile * 16 + m; if (e >= E) e = E - 1;
    const int s = src[e], d = dst[e];
    const float dx = pdst[d * 3 + 0] - psrc[s * 3 + 0];
    const float dy = pdst[d * 3 + 1] - psrc[s * 3 + 1];
    const float dz = pdst[d * 3 + 2] - psrc[s * 3 + 2];
    const float dist = sqrtf(dx * dx + dy * dy + dz * dz);
    v8f acc[8] = {};
    kloop128<NORM>(xdst + (size_t)d * 128, Wa, lane, mi, ri, cb + 256, cb + 384, acc);
    kloop128<NORM>(xsrc + (size_t)s * 128, Wb, lane, mj, rj, cb + 256, cb + 384, acc);
#pragma unroll
    for (int r = 0; r < 8; ++r) {
      const float distr = __shfl(dist, r + hi * 8, 32);
      const int rowg = tile * 16 + r + hi * 8;
      if (rowg < E) {
#pragma unroll
        for (int t = 0; t < 8; ++t) {
          const int col = t * 16 + m;
          float v = acc[t][r] + cb[col] + distr * cb[128 + col];
          hout[(size_t)rowg * 128 + col] = siluf(v);
        }
      }
    }
  }
}

// ---------------------------------------------------------------- generic 128x128
// Y = [silu](X@W + b) [* sigmoid(Y@attw+attb)] [+ resid]   (in-place safe per tile)
template <bool SILU_, bool ATT, bool RESID>
__global__ void k_gemm128(const float* X, int nRows, const float* __restrict__ W,
                          const float* __restrict__ bias, const float* __restrict__ attw,
                          const float* __restrict__ attb, const float* resid, float* Y) {
  extern __shared__ float lds[];
  float* Wl = lds;
  float* bl = lds + 16384;
  float* al = lds + 16512;
  stage_w(W, Wl, 128);
  stage_v(bias, bl, 128);
  if (ATT) stage_v(attw, al, 128);
  __syncthreads();
  const float ab = ATT ? attb[0] : 0.f;
  const int lane = threadIdx.x & 31, hi = lane >> 4, m = lane & 15;
  const int wid = (blockIdx.x * blockDim.x + threadIdx.x) >> 5;
  const int nw = (gridDim.x * blockDim.x) >> 5;
  const int nTiles = (nRows + 15) >> 4;
  for (int tile = wid; tile < nTiles; tile += nw) {
    int rr = tile * 16 + m; if (rr >= nRows) rr = nRows - 1;
    v8f acc[8] = {};
    kloop128<false>(X + (size_t)rr * 128, Wl, lane, 0.f, 1.f, nullptr, nullptr, acc);
#pragma unroll
    for (int t = 0; t < 8; ++t)
#pragma unroll
      for (int r = 0; r < 8; ++r) {
        float v = acc[t][r] + bl[t * 16 + m];
        if (SILU_) v = siluf(v);
        acc[t][r] = v;
      }
    if (ATT) {
#pragma unroll
      for (int r = 0; r < 8; ++r) {
        float p = 0.f;
#pragma unroll
        for (int t = 0; t < 8; ++t) p += acc[t][r] * al[t * 16 + m];
        const float g = sigmf(red16(p) + ab);
#pragma unroll
        for (int t = 0; t < 8; ++t) acc[t][r] *= g;
      }
    }
#pragma unroll
    for (int r = 0; r < 8; ++r) {
      const int rowg = tile * 16 + r + hi * 8;
      if (rowg < nRows) {
#pragma unroll
        for (int t = 0; t < 8; ++t) {
          const int col = t * 16 + m;
          float v = acc[t][r];
          if (RESID) v += resid[(size_t)rowg * 128 + col];
          Y[(size_t)rowg * 128 + col] = v;
        }
      }
    }
  }
}

// ---------------------------------------------------------------- p1 + p2 head
// scal = silu(msg@p1 + b1) @ p2w   (one scalar per edge)
__global__ void k_p12(const float* __restrict__ MSG, int E, const float* __restrict__ W,
                      const float* __restrict__ bias, const float* __restrict__ p2w,
                      float* __restrict__ scal) {
  extern __shared__ float lds[];
  float* Wl = lds;
  float* bl = lds + 16384;
  float* pl = lds + 16512;
  stage_w(W, Wl, 128);
  stage_v(bias, bl, 128);
  stage_v(p2w, pl, 128);
  __syncthreads();
  const int lane = threadIdx.x & 31, hi = lane >> 4, m = lane & 15;
  const int wid = (blockIdx.x * blockDim.x + threadIdx.x) >> 5;
  const int nw = (gridDim.x * blockDim.x) >> 5;
  const int nTiles = (E + 15) >> 4;
  for (int tile = wid; tile < nTiles; tile += nw) {
    int rr = tile * 16 + m; if (rr >= E) rr = E - 1;
    v8f acc[8] = {};
    kloop128<false>(MSG + (size_t)rr * 128, Wl, lane, 0.f, 1.f, nullptr, nullptr, acc);
#pragma unroll
    for (int r = 0; r < 8; ++r) {
      float p = 0.f;
#pragma unroll
      for (int t = 0; t < 8; ++t)
        p += siluf(acc[t][r] + bl[t * 16 + m]) * pl[t * 16 + m];
      const float rs = red16(p);
      const int rowg = tile * 16 + r + hi * 8;
      if (m == 0 && rowg < E) scal[rowg] = rs;
    }
  }
}

// ---------------------------------------------------------------- scatter
template <bool NORM>
__global__ void k_scatter(const float* __restrict__ MSG, const float* __restrict__ scal,
                          const float* __restrict__ psrc, const float* __restrict__ pdst,
                          const int* __restrict__ src, const int* __restrict__ dst, int E,
                          const float* __restrict__ cn_scale, float* __restrict__ agg,
                          float* __restrict__ possum, float* __restrict__ cnt) {
  const int lane = threadIdx.x & 31;
  const int wid = (blockIdx.x * blockDim.x + threadIdx.x) >> 5;
  const int nw = (gridDim.x * blockDim.x) >> 5;
  const float cs = NORM ? cn_scale[0] : 1.f;
  for (int e = wid; e < E; e += nw) {
    const int s = src[e], d = dst[e];
    const float* mr = MSG + (size_t)e * 128 + lane * 4;
    float* ar = agg + (size_t)d * 128 + lane * 4;
#pragma unroll
    for (int i = 0; i < 4; ++i) atomicAdd(ar + i, mr[i]);
    if (lane == 0) {
      float dx = pdst[d * 3 + 0] - psrc[s * 3 + 0];
      float dy = pdst[d * 3 + 1] - psrc[s * 3 + 1];
      float dz = pdst[d * 3 + 2] - psrc[s * 3 + 2];
      const float sc = scal[e];
      if (NORM) {
        float nr = sqrtf(dx * dx + dy * dy + dz * dz);
        nr = fmaxf(nr, 1e-8f);
        const float f = cs / nr * sc;
        dx *= f; dy *= f; dz *= f;
      } else {
        dx *= sc; dy *= sc; dz *= sc;
      }
      atomicAdd(possum + d * 3 + 0, dx);
      atomicAdd(possum + d * 3 + 1, dy);
      atomicAdd(possum + d * 3 + 2, dz);
      atomicAdd(cnt + d, 1.f);
    }
  }
}

__global__ void k_posupd(const float* __restrict__ pd, const float* __restrict__ ps,
                         const float* __restrict__ cnt, int n, float* __restrict__ out) {
  const int i = blockIdx.x * blockDim.x + threadIdx.x;
  if (i < n * 3) out[i] = pd[i] + ps[i] / fmaxf(cnt[i / 3], 1.f);
}

// ---------------------------------------------------------------- u1 (concat GEMM)
__global__ void k_u1(const float* __restrict__ xd, const float* __restrict__ agg, int nRows,
                     const float* __restrict__ W, const float* __restrict__ bias,
                     float* __restrict__ uh) {
  extern __shared__ float lds[];
  float* Wa = lds;
  float* Wb = lds + 16384;
  float* bl = lds + 32768;
  stage_w(W, Wa, 128);
  stage_w(W + 128 * 128, Wb, 128);
  stage_v(bias, bl, 128);
  __syncthreads();
  const int lane = threadIdx.x & 31, hi = lane >> 4, m = lane & 15;
  const int wid = (blockIdx.x * blockDim.x + threadIdx.x) >> 5;
  const int nw = (gridDim.x * blockDim.x) >> 5;
  const int nTiles = (nRows + 15) >> 4;
  for (int tile = wid; tile < nTiles; tile += nw) {
    int rr = tile * 16 + m; if (rr >= nRows) rr = nRows - 1;
    v8f acc[8] = {};
    kloop128<false>(xd + (size_t)rr * 128, Wa, lane, 0.f, 1.f, nullptr, nullptr, acc);
    kloop128<false>(agg + (size_t)rr * 128, Wb, lane, 0.f, 1.f, nullptr, nullptr, acc);
#pragma unroll
    for (int r = 0; r < 8; ++r) {
      const int rowg = tile * 16 + r + hi * 8;
      if (rowg < nRows)
#pragma unroll
        for (int t = 0; t < 8; ++t)
          uh[(size_t)rowg * 128 + t * 16 + m] = siluf(acc[t][r] + bl[t * 16 + m]);
    }
  }
}

// ---------------------------------------------------------------- host side
struct Sub {
  const float *m1w, *m1b, *m2w, *m2b, *p1w, *p1b, *p2w, *u1w, *u1b, *u2w, *u2b;
  const float *attw, *attb, *lnw, *lnb, *cn;
  bool att, norm;
};

// Leaves arrive in JAX tree_leaves order (dict keys sorted): for a2a:
// att{b,w}, m1{b,w}, m2{b,w}, p1{b,w}, p2{w}, u1{b,w}, u2{b,w};
// for a2g/g2a: cn_scale, ln_b, ln_w, m1{b,w}, m2{b,w}, p1{b,w}, p2{w}, u1{b,w}, u2{b,w}
static Sub parse_sub(void* const* din, int& i, bool att, bool norm) {
  Sub s = {};
  s.att = att; s.norm = norm;
  auto F = [&]() { return (const float*)din[i++]; };
  if (att) { s.attb = F(); s.attw = F(); }
  if (norm) { s.cn = F(); s.lnb = F(); s.lnw = F(); }
  s.m1b = F(); s.m1w = F();
  s.m2b = F(); s.m2w = F();
  s.p1b = F(); s.p1w = F();
  s.p2w = F();
  s.u1b = F(); s.u1w = F();
  s.u2b = F(); s.u2w = F();
  return s;
}

static inline int imin(int a, int b) { return a < b ? a : b; }
static inline int imax(int a, int b) { return a > b ? a : b; }

static void run_sublayer(const Sub& S, const float* xsrc, const float* xdst, int n_dst,
                         const float* psrc, const float* pdst, const int* src,
                         const int* dst, int E, float* x_out, float* p_out, float* edge,
                         float* scal, float* agg, float* possum, float* cnt, float* stats,
                         double* lnpart, hipStream_t st) {
  if (S.norm) {
    k_ln_partial<<<LN_BLOCKS, TPB, 0, st>>>(xdst, dst, xsrc, src, E, lnpart);
    k_ln_final<<<1, TPB, 0, st>>>(lnpart, LN_BLOCKS, (long long)E * 128, stats);
  }
  hipMemsetAsync(agg, 0, (size_t)n_dst * 128 * sizeof(float), st);
  hipMemsetAsync(possum, 0, (size_t)n_dst * 3 * sizeof(float), st);
  hipMemsetAsync(cnt, 0, (size_t)n_dst * sizeof(float), st);

  const int eTiles = (E + 15) / 16;
  const int gE = imin(GRID_BIG, imax(1, (eTiles + 7) / 8));
  const size_t lds_m1 = (size_t)(2 * 16384 + 512) * sizeof(float);
  const size_t lds_g = (size_t)(16384 + 256) * sizeof(float);
  const size_t lds_u = (size_t)(2 * 16384 + 128) * sizeof(float);

  if (S.norm)
    k_edge_m1<true><<<gE, TPB, lds_m1, st>>>(xsrc, xdst, psrc, pdst, src, dst, E,
                                             S.m1w, S.m1b, S.lnw, S.lnb, stats, edge);
  else
    k_edge_m1<false><<<gE, TPB, lds_m1, st>>>(xsrc, xdst, psrc, pdst, src, dst, E,
                                              S.m1w, S.m1b, nullptr, nullptr, nullptr, edge);

  if (S.att)
    k_gemm128<true, true, false><<<gE, TPB, lds_g, st>>>(edge, E, S.m2w, S.m2b, S.attw,
                                                         S.attb, nullptr, edge);
  else
    k_gemm128<true, false, false><<<gE, TPB, lds_g, st>>>(edge, E, S.m2w, S.m2b, nullptr,
                                                          nullptr, nullptr, edge);

  k_p12<<<gE, TPB, lds_g, st>>>(edge, E, S.p1w, S.p1b, S.p2w, scal);

  const int gS = imin(2048, imax(1, (E + 7) / 8));
  if (S.norm)
    k_scatter<true><<<gS, TPB, 0, st>>>(edge, scal, psrc, pdst, src, dst, E, S.cn, agg,
                                        possum, cnt);
  else
    k_scatter<false><<<gS, TPB, 0, st>>>(edge, scal, psrc, pdst, src, dst, E, nullptr,
                                         agg, possum, cnt);

  k_posupd<<<imax(1, (n_dst * 3 + TPB - 1) / TPB), TPB, 0, st>>>(pdst, possum, cnt,
                                                                 n_dst, p_out);

  const int nTiles = (n_dst + 15) / 16;
  const int gN = imin(GRID_BIG, imax(1, (nTiles + 7) / 8));
  k_u1<<<gN, TPB, lds_u, st>>>(xdst, agg, n_dst, S.u1w, S.u1b, edge);
  k_gemm128<false, false, true><<<gN, TPB, lds_g, st>>>(edge, n_dst, S.u2w, S.u2b,
                                                        nullptr, nullptr, xdst, x_out);
}

extern "C" void kernel_launch(void* const* d_in, const int* in_sizes, int n_in,
                              void* d_out, int out_size, void* d_ws, size_t ws_size,
                              hipStream_t stream) {
  const float* x_atom0 = (const float*)d_in[0];
  const float* p_atom0 = (const float*)d_in[1];
  const float* x_g0 = (const float*)d_in[2];
  const float* p_g0 = (const float*)d_in[3];
  const int* ei_aa = (const int*)d_in[4];
  const int* ei_ag = (const int*)d_in[5];
  const int* ei_ga = (const int*)d_in[6];
  const int NA = in_sizes[0] / 128;
  const int NG = in_sizes[2] / 128;
  const int EAA = in_sizes[4] / 2;
  const int EAG = in_sizes[5] / 2;
  const int EGA = in_sizes[6] / 2;
  int EMAX = EAA > EAG ? EAA : EAG;
  if (EGA > EMAX) EMAX = EGA;

  int idx = 7;
  Sub L[2][3];
  for (int l = 0; l < 2; ++l) {
    L[l][0] = parse_sub(d_in, idx, true, false);   // a2a
    L[l][1] = parse_sub(d_in, idx, false, true);   // a2g
    L[l][2] = parse_sub(d_in, idx, false, true);   // g2a
  }

  char* cur = (char*)d_ws;
  auto alloc = [&](size_t bytes) -> void* {
    void* p = (void*)cur;
    cur += (bytes + 255) & ~(size_t)255;
    return p;
  };
  float* xA[2] = {(float*)alloc((size_t)NA * 128 * 4), (float*)alloc((size_t)NA * 128 * 4)};
  float* pA[2] = {(float*)alloc((size_t)NA * 3 * 4), (float*)alloc((size_t)NA * 3 * 4)};
  float* xG[2] = {(float*)alloc((size_t)NG * 128 * 4), (float*)alloc((size_t)NG * 128 * 4)};
  float* pG[2] = {(float*)alloc((size_t)NG * 3 * 4), (float*)alloc((size_t)NG * 3 * 4)};
  float* edge = (float*)alloc((size_t)EMAX * 128 * 4);
  float* scal = (float*)alloc((size_t)EMAX * 4);
  float* agg = (float*)alloc((size_t)NA * 128 * 4);
  float* possum = (float*)alloc((size_t)NA * 3 * 4);
  float* cnt = (float*)alloc((size_t)NA * 4);
  float* stats = (float*)alloc(4 * 4);
  double* lnpart = (double*)alloc((size_t)LN_BLOCKS * 4 * 8);

  hipMemcpyAsync(xA[0], x_atom0, (size_t)NA * 128 * 4, hipMemcpyDeviceToDevice, stream);
  hipMemcpyAsync(pA[0], p_atom0, (size_t)NA * 3 * 4, hipMemcpyDeviceToDevice, stream);
  hipMemcpyAsync(xG[0], x_g0, (size_t)NG * 128 * 4, hipMemcpyDeviceToDevice, stream);
  hipMemcpyAsync(pG[0], p_g0, (size_t)NG * 3 * 4, hipMemcpyDeviceToDevice, stream);

  int ca = 0, cg = 0;
  for (int l = 0; l < 2; ++l) {
    run_sublayer(L[l][0], xA[ca], xA[ca], NA, pA[ca], pA[ca], ei_aa, ei_aa + EAA, EAA,
                 xA[1 - ca], pA[1 - ca], edge, scal, agg, possum, cnt, stats, lnpart,
                 stream);
    ca ^= 1;
    run_sublayer(L[l][1], xA[ca], xG[cg], NG, pA[ca], pG[cg], ei_ag, ei_ag + EAG, EAG,
                 xG[1 - cg], pG[1 - cg], edge, scal, agg, possum, cnt, stats, lnpart,
                 stream);
    cg ^= 1;
    run_sublayer(L[l][2], xG[cg], xA[ca], NA, pG[cg], pA[ca], ei_ga, ei_ga + EGA, EGA,
                 xA[1 - ca], pA[1 - ca], edge, scal, agg, possum, cnt, stats, lnpart,
                 stream);
    ca ^= 1;
  }

  float* out = (float*)d_out;
  size_t o = 0;
  hipMemcpyAsync(out + o, xA[ca], (size_t)NA * 128 * 4, hipMemcpyDeviceToDevice, stream);
  o += (size_t)NA * 128;
  hipMemcpyAsync(out + o, xG[cg], (size_t)NG * 128 * 4, hipMemcpyDeviceToDevice, stream);
  o += (size_t)NG * 128;
  hipMemcpyAsync(out + o, pA[ca], (size_t)NA * 3 * 4, hipMemcpyDeviceToDevice, stream);
  o += (size_t)NA * 3;
  hipMemcpyAsync(out + o, pG[cg], (size_t)NG * 3 * 4, hipMemcpyDeviceToDevice, stream);
}